// Generator_40973988004008
// MI455X (gfx1250) — compile-verified
//
#include <hip/hip_runtime.h>
#include <math.h>

typedef __attribute__((ext_vector_type(2))) float v2f;
typedef __attribute__((ext_vector_type(8))) float v8f;

#define B_ROWS 16384
#define LAT    128
#define COND   100
#define D_IN   228          // LAT + COND
#define H      256
#define NC     100
#define ND     100
#define M_MODES 10
#define D_OUT  2100         // NC*(1+M) + ND*M
#define EPSV   1e-5f

// fp32 tensor op: D(16x16) = A(16x4) * B(4x16) + C   -> v_wmma_f32_16x16x4_f32
__device__ __forceinline__ v8f wmma_f32(v2f a, v2f b, v8f c) {
  return __builtin_amdgcn_wmma_f32_16x16x4_f32(false, a, false, b, (short)0, c,
                                               false, false);
}

// A-fragment lane layout (ISA 7.12.2, 32-bit A 16x4): lanes 0-15 hold K=0,1
// (VGPR0,1), lanes 16-31 hold K=2,3.  B (4x16): VGPR0 = rows K=0 (lanes 0-15)
// / K=2 (lanes 16-31), VGPR1 = rows K=1 / K=3.  C/D: VGPR r = row r (lanes
// 0-15) / row 8+r (lanes 16-31), col = lane%16.

template <int MT>
__device__ __forceinline__ void gemm_step(const float* __restrict__ bp0,
                                          const float* __restrict__ bp1,
                                          int n0, int l16, const v2f (&a)[MT],
                                          v8f (&acc)[MT][4]) {
#pragma unroll
  for (int nt = 0; nt < 4; ++nt) {
    v2f bb;
    bb.x = bp0[n0 + nt * 16 + l16];
    bb.y = bp1[n0 + nt * 16 + l16];
#pragma unroll
    for (int mt = 0; mt < MT; ++mt)
      acc[mt][nt] = wmma_f32(a[mt], bb, acc[mt][nt]);
  }
}

// ---------------- GEMM1: h1 = [z|c] @ W1 + b1  (wave tile 32x64) -------------
__global__ __launch_bounds__(128) void k_gemm1(
    const float* __restrict__ z, const float* __restrict__ c,
    const float* __restrict__ W1, const float* __restrict__ b1,
    float* __restrict__ h1) {
  constexpr int MT = 2;                         // 32 rows per wave
  const int lane = threadIdx.x & 31;
  const int wave = threadIdx.x >> 5;
  const int l16  = lane & 15;
  const int hi   = lane >> 4;
  const int m0   = blockIdx.x * (MT * 16);
  const int n0   = wave * 64;

  const float* zrow[MT];
  const float* crow[MT];
#pragma unroll
  for (int mt = 0; mt < MT; ++mt) {
    zrow[mt] = z + (size_t)(m0 + mt * 16 + l16) * LAT + 2 * hi;
    crow[mt] = c + (size_t)(m0 + mt * 16 + l16) * COND + 2 * hi;
  }

  v8f acc[MT][4];
#pragma unroll
  for (int mt = 0; mt < MT; ++mt)
#pragma unroll
    for (int nt = 0; nt < 4; ++nt) acc[mt][nt] = (v8f){};

  // segment 1: K in [0,128) from z  (branch-free)
  for (int k = 0; k < LAT; k += 4) {
    v2f a[MT];
#pragma unroll
    for (int mt = 0; mt < MT; ++mt) a[mt] = *(const v2f*)(zrow[mt] + k);
    const float* bp0 = W1 + (size_t)(k + 2 * hi) * H;
    gemm_step<MT>(bp0, bp0 + H, n0, l16, a, acc);
  }
  // segment 2: K in [128,228) from c  (branch-free)
  for (int k = 0; k < COND; k += 4) {
    v2f a[MT];
#pragma unroll
    for (int mt = 0; mt < MT; ++mt) a[mt] = *(const v2f*)(crow[mt] + k);
    const float* bp0 = W1 + (size_t)(LAT + k + 2 * hi) * H;
    gemm_step<MT>(bp0, bp0 + H, n0, l16, a, acc);
  }

#pragma unroll
  for (int mt = 0; mt < MT; ++mt)
#pragma unroll
    for (int nt = 0; nt < 4; ++nt) {
      const int n = n0 + nt * 16 + l16;
      const float bias = b1[n];
#pragma unroll
      for (int r = 0; r < 8; ++r)
        h1[(size_t)(m0 + mt * 16 + hi * 8 + r) * H + n] = acc[mt][nt][r] + bias;
    }
}

// ------------- deterministic column stats (sum / sum of squares) -------------
__global__ __launch_bounds__(256) void k_colstats(const float* __restrict__ h,
                                                  float* __restrict__ psum,
                                                  float* __restrict__ psq) {
  const int j = threadIdx.x;                    // column 0..255
  const int b = blockIdx.x;                     // row chunk 0..255 (64 rows)
  const float* p = h + (size_t)b * 64 * H + j;
  float s = 0.f, q = 0.f;
  for (int r = 0; r < 64; ++r) {
    float v = p[(size_t)r * H];
    s += v;
    q = fmaf(v, v, q);
  }
  psum[b * H + j] = s;
  psq[b * H + j]  = q;
}

// fold BN into per-column affine: o = relu(h*a + cc)
__global__ __launch_bounds__(256) void k_fold(const float* __restrict__ psum,
                                              const float* __restrict__ psq,
                                              const float* __restrict__ g,
                                              const float* __restrict__ be,
                                              float* __restrict__ a,
                                              float* __restrict__ cc) {
  const int j = threadIdx.x;
  float s = 0.f, q = 0.f;
  for (int b = 0; b < 256; ++b) { s += psum[b * H + j]; q += psq[b * H + j]; }
  const float mu  = s * (1.f / (float)B_ROWS);
  const float var = q * (1.f / (float)B_ROWS) - mu * mu;
  const float rs  = rsqrtf(var + EPSV);
  const float av  = g[j] * rs;
  a[j]  = av;
  cc[j] = be[j] - mu * av;
}

// ------- GEMM2: h2 = [relu(bn(h1)) | z | c] @ W2 + b2  (wave tile 32x64) -----
__global__ __launch_bounds__(128) void k_gemm2(
    const float* __restrict__ z, const float* __restrict__ c,
    const float* __restrict__ h1, const float* __restrict__ a1,
    const float* __restrict__ c1, const float* __restrict__ W2,
    const float* __restrict__ b2, float* __restrict__ h2) {
  constexpr int MT = 2;
  const int lane = threadIdx.x & 31;
  const int wave = threadIdx.x >> 5;
  const int l16  = lane & 15;
  const int hi   = lane >> 4;
  const int m0   = blockIdx.x * (MT * 16);
  const int n0   = wave * 64;

  const float* hrow[MT];
  const float* zrow[MT];
  const float* crow[MT];
#pragma unroll
  for (int mt = 0; mt < MT; ++mt) {
    hrow[mt] = h1 + (size_t)(m0 + mt * 16 + l16) * H + 2 * hi;
    zrow[mt] = z + (size_t)(m0 + mt * 16 + l16) * LAT + 2 * hi;
    crow[mt] = c + (size_t)(m0 + mt * 16 + l16) * COND + 2 * hi;
  }

  v8f acc[MT][4];
#pragma unroll
  for (int mt = 0; mt < MT; ++mt)
#pragma unroll
    for (int nt = 0; nt < 4; ++nt) acc[mt][nt] = (v8f){};

  // segment 1: K in [0,256) = relu(bn(h1))  (branch-free)
  for (int k = 0; k < H; k += 4) {
    const v2f av = *(const v2f*)(a1 + k + 2 * hi);  // BN affine shared by mt
    const v2f cv = *(const v2f*)(c1 + k + 2 * hi);
    v2f a[MT];
#pragma unroll
    for (int mt = 0; mt < MT; ++mt) {
      v2f hv = *(const v2f*)(hrow[mt] + k);
      a[mt].x = fmaxf(fmaf(hv.x, av.x, cv.x), 0.f);
      a[mt].y = fmaxf(fmaf(hv.y, av.y, cv.y), 0.f);
    }
    const float* bp0 = W2 + (size_t)(k + 2 * hi) * H;
    gemm_step<MT>(bp0, bp0 + H, n0, l16, a, acc);
  }
  // segment 2: K in [256,384) from z
  for (int k = 0; k < LAT; k += 4) {
    v2f a[MT];
#pragma unroll
    for (int mt = 0; mt < MT; ++mt) a[mt] = *(const v2f*)(zrow[mt] + k);
    const float* bp0 = W2 + (size_t)(H + k + 2 * hi) * H;
    gemm_step<MT>(bp0, bp0 + H, n0, l16, a, acc);
  }
  // segment 3: K in [384,484) from c
  for (int k = 0; k < COND; k += 4) {
    v2f a[MT];
#pragma unroll
    for (int mt = 0; mt < MT; ++mt) a[mt] = *(const v2f*)(crow[mt] + k);
    const float* bp0 = W2 + (size_t)(H + LAT + k + 2 * hi) * H;
    gemm_step<MT>(bp0, bp0 + H, n0, l16, a, acc);
  }

#pragma unroll
  for (int mt = 0; mt < MT; ++mt)
#pragma unroll
    for (int nt = 0; nt < 4; ++nt) {
      const int n = n0 + nt * 16 + l16;
      const float bias = b2[n];
#pragma unroll
      for (int r = 0; r < 8; ++r)
        h2[(size_t)(m0 + mt * 16 + hi * 8 + r) * H + n] = acc[mt][nt][r] + bias;
    }
}

// ---- GEMM3: x = relu(bn(h2)) @ Wout + bout (into d_out), wave tile 64x64 ----
__global__ __launch_bounds__(128) void k_gemm3(
    const float* __restrict__ h2, const float* __restrict__ a2,
    const float* __restrict__ c2, const float* __restrict__ Wout,
    const float* __restrict__ bout, float* __restrict__ x) {
  constexpr int MT = 4;                         // 64 rows per wave
  const int lane = threadIdx.x & 31;
  const int wave = threadIdx.x >> 5;
  const int l16  = lane & 15;
  const int hi   = lane >> 4;
  const int m0   = blockIdx.x * (MT * 16);
  const int n0   = blockIdx.y * 256 + wave * 64;
  if (n0 >= D_OUT) return;                      // wave-uniform exit, no barriers

  // Per-tile clamped column + mask (keeps EXEC uniform around WMMA).
  int   nn[4];
  int   ncl[4];
  float msk[4];
#pragma unroll
  for (int t = 0; t < 4; ++t) {
    nn[t]  = n0 + t * 16 + l16;
    ncl[t] = nn[t] < D_OUT ? nn[t] : (D_OUT - 1);
    msk[t] = nn[t] < D_OUT ? 1.f : 0.f;
  }

  const float* hrow[MT];
#pragma unroll
  for (int mt = 0; mt < MT; ++mt)
    hrow[mt] = h2 + (size_t)(m0 + mt * 16 + l16) * H + 2 * hi;

  v8f acc[MT][4];
#pragma unroll
  for (int mt = 0; mt < MT; ++mt)
#pragma unroll
    for (int nt = 0; nt < 4; ++nt) acc[mt][nt] = (v8f){};

  for (int k = 0; k < H; k += 4) {
    const v2f av = *(const v2f*)(a2 + k + 2 * hi);
    const v2f cv = *(const v2f*)(c2 + k + 2 * hi);
    v2f a[MT];
#pragma unroll
    for (int mt = 0; mt < MT; ++mt) {
      v2f hv = *(const v2f*)(hrow[mt] + k);
      a[mt].x = fmaxf(fmaf(hv.x, av.x, cv.x), 0.f);
      a[mt].y = fmaxf(fmaf(hv.y, av.y, cv.y), 0.f);
    }
    const float* bp0 = Wout + (size_t)(k + 2 * hi) * D_OUT;
    const float* bp1 = bp0 + D_OUT;
#pragma unroll
    for (int nt = 0; nt < 4; ++nt) {
      v2f bb;
      bb.x = bp0[ncl[nt]] * msk[nt];
      bb.y = bp1[ncl[nt]] * msk[nt];
#pragma unroll
      for (int mt = 0; mt < MT; ++mt)
        acc[mt][nt] = wmma_f32(a[mt], bb, acc[mt][nt]);
    }
  }

#pragma unroll
  for (int mt = 0; mt < MT; ++mt)
#pragma unroll
    for (int nt = 0; nt < 4; ++nt) {
      if (nn[nt] < D_OUT) {
        const float bias = bout[nn[nt]];
#pragma unroll
        for (int r = 0; r < 8; ++r)
          x[(size_t)(m0 + mt * 16 + hi * 8 + r) * D_OUT + nn[nt]] =
              acc[mt][nt][r] + bias;
      }
    }
}

// ---------------- epilogue: tanh + hard gumbel-softmax == one-hot(argmax) ---
__global__ __launch_bounds__(256) void k_epilogue(
    float* __restrict__ x, const float* __restrict__ gum_c,
    const float* __restrict__ gum_d) {
  const int idx = blockIdx.x * 256 + threadIdx.x;
  if (idx >= B_ROWS * (NC + ND)) return;
  const int row = idx / (NC + ND);
  const int grp = idx - row * (NC + ND);
  if (grp < NC) {
    float* p = x + (size_t)row * D_OUT + grp * (1 + M_MODES);
    p[0] = tanhf(p[0]);
    const float* g = gum_c + ((size_t)row * NC + grp) * M_MODES;
    int amax = 0;
    float best = p[1] + __builtin_nontemporal_load(g);
#pragma unroll
    for (int j = 1; j < M_MODES; ++j) {
      float v = p[1 + j] + __builtin_nontemporal_load(g + j);
      if (v > best) { best = v; amax = j; }     // first-max tie-break like argmax
    }
#pragma unroll
    for (int j = 0; j < M_MODES; ++j) p[1 + j] = (j == amax) ? 1.f : 0.f;
  } else {
    const int gd = grp - NC;
    float* p = x + (size_t)row * D_OUT + NC * (1 + M_MODES) + gd * M_MODES;
    const float* g = gum_d + ((size_t)row * ND + gd) * M_MODES;
    int amax = 0;
    float best = p[0] + __builtin_nontemporal_load(g);
#pragma unroll
    for (int j = 1; j < M_MODES; ++j) {
      float v = p[j] + __builtin_nontemporal_load(g + j);
      if (v > best) { best = v; amax = j; }
    }
#pragma unroll
    for (int j = 0; j < M_MODES; ++j) p[j] = (j == amax) ? 1.f : 0.f;
  }
}

extern "C" void kernel_launch(void* const* d_in, const int* in_sizes, int n_in,
                              void* d_out, int out_size, void* d_ws,
                              size_t ws_size, hipStream_t stream) {
  (void)in_sizes; (void)n_in; (void)out_size; (void)ws_size;
  const float* z    = (const float*)d_in[0];
  const float* c    = (const float*)d_in[1];
  const float* W1   = (const float*)d_in[2];
  const float* b1   = (const float*)d_in[3];
  const float* g1   = (const float*)d_in[4];
  const float* be1  = (const float*)d_in[5];
  const float* W2   = (const float*)d_in[6];
  const float* b2   = (const float*)d_in[7];
  const float* g2   = (const float*)d_in[8];
  const float* be2  = (const float*)d_in[9];
  const float* Wout = (const float*)d_in[10];
  const float* bout = (const float*)d_in[11];
  const float* gumc = (const float*)d_in[12];
  const float* gumd = (const float*)d_in[13];
  float* out = (float*)d_out;

  // workspace layout (floats): h1[B*H] h2[B*H] psum[256*H] psq[256*H]
  //                            a1[H] c1[H] a2[H] c2[H]   (~34 MB total)
  float* ws   = (float*)d_ws;
  float* h1   = ws;
  float* h2   = h1 + (size_t)B_ROWS * H;
  float* psum = h2 + (size_t)B_ROWS * H;
  float* psq  = psum + 256 * H;
  float* a1   = psq + 256 * H;
  float* c1   = a1 + H;
  float* a2   = c1 + H;
  float* c2   = a2 + H;

  const dim3 blk(128);
  k_gemm1<<<dim3(B_ROWS / 32), blk, 0, stream>>>(z, c, W1, b1, h1);
  k_colstats<<<256, 256, 0, stream>>>(h1, psum, psq);
  k_fold<<<1, 256, 0, stream>>>(psum, psq, g1, be1, a1, c1);
  k_gemm2<<<dim3(B_ROWS / 32), blk, 0, stream>>>(z, c, h1, a1, c1, W2, b2, h2);
  k_colstats<<<256, 256, 0, stream>>>(h2, psum, psq);
  k_fold<<<1, 256, 0, stream>>>(psum, psq, g2, be2, a2, c2);
  k_gemm3<<<dim3(B_ROWS / 64, 9), blk, 0, stream>>>(h2, a2, c2, Wout, bout, out);
  const int ngrp = B_ROWS * (NC + ND);
  k_epilogue<<<(ngrp + 255) / 256, 256, 0, stream>>>(out, gumc, gumd);
}